// HNHN_62242666053887
// MI455X (gfx1250) — compile-verified
//
#include <hip/hip_runtime.h>
#include <math.h>

// ---------------- problem constants (match reference) ----------------
static constexpr int NV   = 100000;
static constexpr int NE   = 20000;
static constexpr int NNZ  = 3200000;
static constexpr int CH   = 128;   // in_ch == n_hid
static constexpr int NCLS = 40;
static constexpr int NCP  = 48;    // class dim padded to 3x16 WMMA tiles
static constexpr float BN_EPS = 1e-5f;

typedef __attribute__((ext_vector_type(2))) float v2f;
typedef __attribute__((ext_vector_type(8))) float v8f;

// ---------------- utility kernels ----------------
__global__ void zero_f32(float* __restrict__ p, int n) {
    int i = blockIdx.x * blockDim.x + threadIdx.x;
    if (i < n) p[i] = 0.0f;
}

__global__ void zero_i32(int* __restrict__ p, int n) {
    int i = blockIdx.x * blockDim.x + threadIdx.x;
    if (i < n) p[i] = 0;
}

__global__ void count_kernel(const int* __restrict__ e_ids,
                             const int* __restrict__ v_ids,
                             int* __restrict__ cnt_e, int* __restrict__ cnt_v, int nnz) {
    int i = blockIdx.x * blockDim.x + threadIdx.x;
    if (i < nnz) {
        atomicAdd(&cnt_e[e_ids[i]], 1);
        atomicAdd(&cnt_v[v_ids[i]], 1);
    }
}

__global__ void inv_kernel(const int* __restrict__ cnt, float* __restrict__ inv, int n) {
    int i = blockIdx.x * blockDim.x + threadIdx.x;
    if (i < n) inv[i] = 1.0f / fmaxf((float)cnt[i], 1.0f);
}

// ---------------- WMMA fp32 GEMM: C[M x N] = op(A[M x KK]) * B[KK x Nvalid] ----------------
// One block = one 16-row M tile; one wave = one 16-col N tile.
// The A tile is staged through LDS once per block with the segment-mean scale and
// ReLU fused in (RSCALE), so the K loop is pure ds_load_b64 + global b32 + v_wmma.
// MASKN: zero-fill B columns >= Nvalid via clamped index + 0/1 mask (branch-free),
// so padded output columns come out exactly 0.
template <int KK, bool RSCALE, bool MASKN>
__global__ void gemm_wmma_f32(const float* __restrict__ A, int lda,
                              const float* __restrict__ B, int ldb,
                              float* __restrict__ C, int ldc,
                              int Nvalid, const float* __restrict__ scaleA) {
    constexpr int LDSROW = KK + 4;          // pad: bank = k + 4*lm (mod 64), conflict-free
    __shared__ float As[16 * LDSROW];
    const int row0 = blockIdx.x << 4;

    // cooperative stage of the 16 x KK A tile
    for (int i = threadIdx.x; i < 16 * KK; i += blockDim.x) {
        const int r = i / KK;
        const int k = i - r * KK;
        float v = A[(size_t)(row0 + r) * lda + k];
        if (RSCALE) v = fmaxf(v * scaleA[row0 + r], 0.0f);
        As[r * LDSROW + k] = v;
    }
    __syncthreads();

    const int wave = threadIdx.x >> 5;      // N-tile
    const int lane = threadIdx.x & 31;
    const int lm = lane & 15;
    const int lh = lane >> 4;               // half-wave: selects K pair within step of 4
    const int n0 = wave << 4;

    int   ncol = n0 + lm;                   // B column for this lane
    float bm   = 1.0f;
    if (MASKN) {
        bm   = (ncol < Nvalid) ? 1.0f : 0.0f;
        ncol = min(ncol, Nvalid - 1);
    }
    const float* Alane = As + lm * LDSROW + lh * 2;
    const float* Bcol  = B + ncol + (size_t)(lh * 2) * ldb;

    v8f c = {};
#pragma unroll 4
    for (int kb = 0; kb < KK; kb += 4) {
        // A frag: lane holds A[m][kb+lh*2], A[m][kb+lh*2+1]  (8B-aligned LDS read)
        v2f a = *(const v2f*)(Alane + kb);
        // B frag: lane holds B[kb+lh*2][ncol], B[kb+lh*2+1][ncol]
        v2f b;
        b[0] = Bcol[(size_t)kb * ldb] * bm;
        b[1] = Bcol[(size_t)(kb + 1) * ldb] * bm;
        c = __builtin_amdgcn_wmma_f32_16x16x4_f32(false, a, false, b, (short)0, c,
                                                  false, false);
    }
    // C/D layout: VGPR r -> row r (lanes 0-15) / row r+8 (lanes 16-31), col = lane%16
    float* Cbase = C + (size_t)row0 * ldc + n0;
#pragma unroll
    for (int r = 0; r < 8; ++r)
        Cbase[(size_t)(r + lh * 8) * ldc + lm] = c[r];
}

// ---------------- scatter-add (segment-sum building block) ----------------
// TPE threads cooperate on one nnz entry, each moving a float4 of channels.
template <int TPE>
__global__ void scatter_add(const float* __restrict__ src, int lds,
                            float* __restrict__ dst, int ldd,
                            const int* __restrict__ sidx,
                            const int* __restrict__ didx, int nnz) {
    const int epb = blockDim.x / TPE;
    const int e = blockIdx.x * epb + (int)(threadIdx.x / TPE);
    if (e >= nnz) return;
    const int cg = (threadIdx.x % TPE) * 4;
    const int sr = sidx[e];
    const int dr = didx[e];
    const float4 v = *(const float4*)(src + (size_t)sr * lds + cg);
    float* d = dst + (size_t)dr * ldd + cg;
    unsafeAtomicAdd(d + 0, v.x);   // global_atomic_add_f32, no return
    unsafeAtomicAdd(d + 1, v.y);
    unsafeAtomicAdd(d + 2, v.z);
    unsafeAtomicAdd(d + 3, v.w);
}

// ---------------- fused mean + ReLU + BatchNorm (eval) ----------------
__global__ void bn_relu_kernel(float* __restrict__ h, const float* __restrict__ inv_v,
                               const float* __restrict__ gamma, const float* __restrict__ beta,
                               const float* __restrict__ mean,  const float* __restrict__ var,
                               int total) {
    int i = blockIdx.x * blockDim.x + threadIdx.x;
    if (i >= total) return;
    const int c = i & (CH - 1);
    const int r = i >> 7;
    float x = fmaxf(h[i] * inv_v[r], 0.0f);
    h[i] = (x - mean[c]) * rsqrtf(var[c] + BN_EPS) * gamma[c] + beta[c];
}

// ---------------- fused mean + log_softmax over 40 classes ----------------
__global__ void lsm_kernel(const float* __restrict__ acc, const float* __restrict__ inv_v,
                           float* __restrict__ out, int nV) {
    int r = blockIdx.x * blockDim.x + threadIdx.x;
    if (r >= nV) return;
    const float* row = acc + (size_t)r * NCP;
    const float s = inv_v[r];
    float v[NCLS];
    float mx = -INFINITY;
#pragma unroll
    for (int c = 0; c < NCLS; ++c) { v[c] = row[c] * s; mx = fmaxf(mx, v[c]); }
    float sum = 0.0f;
#pragma unroll
    for (int c = 0; c < NCLS; ++c) sum += __expf(v[c] - mx);
    const float l = __logf(sum) + mx;
    float* orow = out + (size_t)r * NCLS;
#pragma unroll
    for (int c = 0; c < NCLS; ++c) orow[c] = v[c] - l;
}

// ---------------- driver ----------------
extern "C" void kernel_launch(void* const* d_in, const int* in_sizes, int n_in,
                              void* d_out, int out_size, void* d_ws, size_t ws_size,
                              hipStream_t stream) {
    const float* x       = (const float*)d_in[0];
    const int*   v_ids   = (const int*)d_in[1];
    const int*   e_ids   = (const int*)d_in[2];
    const float* w1_v2e  = (const float*)d_in[3];   // 128x128, row-major (in,out)
    const float* w1_e2v  = (const float*)d_in[4];   // 128x128
    const float* w2_v2e  = (const float*)d_in[5];   // 128x40
    const float* w2_e2v  = (const float*)d_in[6];   // 40x40
    const float* bn_g    = (const float*)d_in[7];
    const float* bn_b    = (const float*)d_in[8];
    const float* bn_m    = (const float*)d_in[9];
    const float* bn_v    = (const float*)d_in[10];
    float* out = (float*)d_out;

    // workspace layout (floats)
    float* ws = (float*)d_ws;
    float* bufV1 = ws;                                   // NV*128: x@W1 -> e2v acc/h -> final acc (NV*48)
    float* bufE1 = bufV1 + (size_t)NV * CH;              // NE*128: v2e accumulators (both layers)
    float* bufE2 = bufE1 + (size_t)NE * CH;              // NE*128: y=GEMM outputs (both layers)
    float* bufV2 = bufE2 + (size_t)NE * CH;              // NV*48 : h@W2
    int*   cnt_e = (int*)(bufV2 + (size_t)NV * NCP);     // NE
    int*   cnt_v = cnt_e + NE;                           // NV
    float* inv_e = (float*)(cnt_v + NV);                 // NE
    float* inv_v = inv_e + NE;                           // NV

    const int TB = 256;
    auto cdiv = [](long long a, long long b) { return (int)((a + b - 1) / b); };

    // --- degrees -> reciprocals (shared by both layers) ---
    zero_i32<<<cdiv(NE + NV, TB), TB, 0, stream>>>(cnt_e, NE + NV);
    count_kernel<<<cdiv(NNZ, TB), TB, 0, stream>>>(e_ids, v_ids, cnt_e, cnt_v, NNZ);
    inv_kernel<<<cdiv(NE, TB), TB, 0, stream>>>(cnt_e, inv_e, NE);
    inv_kernel<<<cdiv(NV, TB), TB, 0, stream>>>(cnt_v, inv_v, NV);

    // ================= layer 1 (C = 128) =================
    // xw = x @ w1_v2e            (NV x 128)
    gemm_wmma_f32<CH, false, false><<<NV / 16, 8 * 32, 0, stream>>>(
        x, CH, w1_v2e, CH, bufV1, CH, CH, nullptr);
    // v2e segment-sum into bufE1
    zero_f32<<<cdiv((size_t)NE * CH, TB), TB, 0, stream>>>(bufE1, NE * CH);
    scatter_add<32><<<cdiv(NNZ, 8), 256, 0, stream>>>(bufV1, CH, bufE1, CH, v_ids, e_ids, NNZ);
    // y = relu(sum * inv_e) @ w1_e2v   (mean+relu fused into the LDS stage)
    gemm_wmma_f32<CH, true, false><<<NE / 16, 8 * 32, 0, stream>>>(
        bufE1, CH, w1_e2v, CH, bufE2, CH, CH, inv_e);
    // e2v segment-sum into bufV1
    zero_f32<<<cdiv((size_t)NV * CH, TB), TB, 0, stream>>>(bufV1, NV * CH);
    scatter_add<32><<<cdiv(NNZ, 8), 256, 0, stream>>>(bufE2, CH, bufV1, CH, e_ids, v_ids, NNZ);
    // h = BN(relu(sum * inv_v)) in place
    bn_relu_kernel<<<cdiv((size_t)NV * CH, TB), TB, 0, stream>>>(bufV1, inv_v, bn_g, bn_b,
                                                                 bn_m, bn_v, NV * CH);

    // ================= layer 2 (C = 40, padded to 48) =================
    // hw = h @ w2_v2e            (NV x 48, cols 40..47 == 0)
    gemm_wmma_f32<CH, false, true><<<NV / 16, 3 * 32, 0, stream>>>(
        bufV1, CH, w2_v2e, NCLS, bufV2, NCP, NCLS, nullptr);
    // v2e segment-sum into bufE1 (stride 48, move the 40 valid channels)
    zero_f32<<<cdiv((size_t)NE * NCP, TB), TB, 0, stream>>>(bufE1, NE * NCP);
    scatter_add<10><<<cdiv(NNZ, 24), 240, 0, stream>>>(bufV2, NCP, bufE1, NCP, v_ids, e_ids, NNZ);
    // y2 = relu(sum * inv_e) @ w2_e2v   (K = 40)
    gemm_wmma_f32<NCLS, true, true><<<NE / 16, 3 * 32, 0, stream>>>(
        bufE1, NCP, w2_e2v, NCLS, bufE2, NCP, NCLS, inv_e);
    // e2v segment-sum into bufV1 (stride 48)
    zero_f32<<<cdiv((size_t)NV * NCP, TB), TB, 0, stream>>>(bufV1, NV * NCP);
    scatter_add<10><<<cdiv(NNZ, 24), 240, 0, stream>>>(bufE2, NCP, bufV1, NCP, e_ids, v_ids, NNZ);
    // out = log_softmax(sum * inv_v)  -> (NV x 40)
    lsm_kernel<<<cdiv(NV, TB), TB, 0, stream>>>(bufV1, inv_v, out, NV);
}